// HMM_45844480918010
// MI455X (gfx1250) — compile-verified
//
#include <hip/hip_runtime.h>
#include <cstdint>

#define HMM_B 64
#define HMM_T 1024
#define HMM_S 512

static constexpr float ONE_MINUS_EPS = 0.9f;                 // 1 - eps
static constexpr float LEAK          = 0.1f / (float)HMM_S;  // eps / S
static constexpr double LN_2PI       = 1.8378770664093454;

typedef float v2f __attribute__((ext_vector_type(2)));
typedef float v8f __attribute__((ext_vector_type(8)));

// ---- VALU-only wave32 butterflies: 4x DPP ROW_XMASK + 1x permlanex16 ----
__device__ __forceinline__ float perm_x16(float x) {
  int i = __builtin_amdgcn_permlanex16(__float_as_int(x), __float_as_int(x),
                                       0x76543210, 0xfedcba98, false, false);
  return __int_as_float(i);
}
__device__ __forceinline__ float wave_sum(float x) {
  int i;
  i = __builtin_amdgcn_mov_dpp(__float_as_int(x), 0x161, 0xf, 0xf, true);  // xor 1
  x += __int_as_float(i);
  i = __builtin_amdgcn_mov_dpp(__float_as_int(x), 0x162, 0xf, 0xf, true);  // xor 2
  x += __int_as_float(i);
  i = __builtin_amdgcn_mov_dpp(__float_as_int(x), 0x164, 0xf, 0xf, true);  // xor 4
  x += __int_as_float(i);
  i = __builtin_amdgcn_mov_dpp(__float_as_int(x), 0x168, 0xf, 0xf, true);  // xor 8
  x += __int_as_float(i);
  x += perm_x16(x);                                                        // xor 16
  return x;
}
__device__ __forceinline__ float wave_max(float x) {
  int i;
  i = __builtin_amdgcn_mov_dpp(__float_as_int(x), 0x161, 0xf, 0xf, true);
  x = fmaxf(x, __int_as_float(i));
  i = __builtin_amdgcn_mov_dpp(__float_as_int(x), 0x162, 0xf, 0xf, true);
  x = fmaxf(x, __int_as_float(i));
  i = __builtin_amdgcn_mov_dpp(__float_as_int(x), 0x164, 0xf, 0xf, true);
  x = fmaxf(x, __int_as_float(i));
  i = __builtin_amdgcn_mov_dpp(__float_as_int(x), 0x168, 0xf, 0xf, true);
  x = fmaxf(x, __int_as_float(i));
  x = fmaxf(x, perm_x16(x));
  return x;
}

// Depth-4 pairwise tree over 16 lane-local values (fixed order -> deterministic)
__device__ __forceinline__ float tree_sum16(const float* u) {
  float a0 = u[0] + u[1],   a1 = u[2] + u[3];
  float a2 = u[4] + u[5],   a3 = u[6] + u[7];
  float a4 = u[8] + u[9],   a5 = u[10] + u[11];
  float a6 = u[12] + u[13], a7 = u[14] + u[15];
  float b0 = a0 + a1, b1 = a2 + a3, b2 = a4 + a5, b3 = a6 + a7;
  float c0 = b0 + b1, c1 = b2 + b3;
  return c0 + c1;
}

// One wave (32 lanes) per batch. 16 states per lane, all register-resident.
// Sum-normalized probability-domain scan: ln(alpha_j) == L + ln(u_j).
// Per step only ONE wave reduction (U = sum u); normalization is folded into
// the scalar coefficient c = 0.9 * rcp(U), so no per-state normalize multiply.
__global__ __launch_bounds__(32) void hmm_scan_kernel(
    const float* __restrict__ obvs, const float* __restrict__ mu,
    const float* __restrict__ ln_pi, double* __restrict__ ws)
{
  __shared__ float s_o[HMM_T];  // 4 KB: this batch's observation row
  const int b    = blockIdx.x;
  const int lane = threadIdx.x;

  // ---- CDNA5 async global->LDS copy of the 4KB observation row ----
  {
    const unsigned long long gbase =
        (unsigned long long)(const void*)(obvs + (size_t)b * HMM_T);
    const unsigned int lds_base = (unsigned int)(uintptr_t)(&s_o[0]);
#pragma unroll
    for (int i = 0; i < 8; ++i) {
      unsigned int voff = (unsigned int)(i * 32 + lane) * 16u;  // 16B per lane
      unsigned int ldso = lds_base + voff;
      asm volatile("global_load_async_to_lds_b128 %0, %1, %2"
                   :: "v"(ldso), "v"(voff), "s"(gbase) : "memory");
    }
    asm volatile("s_wait_asynccnt 0" ::: "memory");
  }

  // State j = lane*16 + k
  float muv[16];
  {
    const float4* mu4 = (const float4*)mu;
#pragma unroll
    for (int i = 0; i < 4; ++i) {
      float4 v = mu4[lane * 4 + i];
      muv[i * 4 + 0] = v.x; muv[i * 4 + 1] = v.y;
      muv[i * 4 + 2] = v.z; muv[i * 4 + 3] = v.w;
    }
  }

  float u[16];   // scaled forward weights, ln(alpha) = L + ln(u)
  double L;      // accumulated log-scale (wave-uniform)
  {
    const float o0 = s_o[0];
    const float4* pi4 = (const float4*)ln_pi;
    float c0[16];
    float mloc = -3.4e38f;
#pragma unroll
    for (int i = 0; i < 4; ++i) {
      float4 v = pi4[lane * 4 + i];
      float pi_[4] = {v.x, v.y, v.z, v.w};
#pragma unroll
      for (int c = 0; c < 4; ++c) {
        int k = i * 4 + c;
        float d = o0 - muv[k];
        c0[k] = pi_[c] - 0.5f * d * d;   // -0.5*ln(2pi) deferred analytically
        mloc = fmaxf(mloc, c0[k]);
      }
    }
    const float M = wave_max(mloc);
#pragma unroll
    for (int k = 0; k < 16; ++k) u[k] = __expf(c0[k] - M);
    L = (double)M;
  }

  // alpha'_j = em_j * (0.9*alpha_j + (eps/S)*Sum(alpha))
  // stored:   u'_j  = em_j * fma(c, u_j, LEAK),  c = 0.9*rcp(U),  L -= log(r)
#pragma unroll 2
  for (int t = 1; t < HMM_T; ++t) {
    const float o = s_o[t];

    const float U = wave_sum(tree_sum16(u));
    const float r = __builtin_amdgcn_rcpf(U);
    const float c = ONE_MINUS_EPS * r;
    L -= (double)logf(r);

#pragma unroll
    for (int k = 0; k < 16; ++k) {
      float d  = o - muv[k];
      float em = __expf(-0.5f * d * d);          // constant factor deferred
      u[k] = fmaf(c, u[k], LEAK) * em;
    }
  }

  const float Utot = wave_sum(tree_sum16(u));
  // T emission factors, each deferred a -0.5*ln(2pi) constant
  const double res = L + (double)logf(Utot) - 0.5 * LN_2PI * (double)HMM_T;
  if (lane == 0) ws[b] = res;
}

// Deterministic 64-way reduction using V_WMMA_F32_16X16X4_F32:
// values laid out as A (16x4, ISA layout), B = ones => D rows hold row-sums.
__global__ __launch_bounds__(32) void hmm_reduce_kernel(
    const double* __restrict__ ws, float* __restrict__ out)
{
  const int lane = threadIdx.x;
  const int m  = lane & 15;            // A-matrix row
  const int kb = (lane >> 4) << 1;     // lanes 0-15: K=0,1 ; lanes 16-31: K=2,3

  v2f a;
  a.x = (float)ws[m * 4 + kb + 0];
  a.y = (float)ws[m * 4 + kb + 1];
  v2f ones; ones.x = 1.0f; ones.y = 1.0f;
  v8f c = {};
  c = __builtin_amdgcn_wmma_f32_16x16x4_f32(
      /*neg_a=*/false, a, /*neg_b=*/false, ones,
      /*c_mod=*/(short)0, c, /*reuse_a=*/false, /*reuse_b=*/false);

  // Each lane's 8 accumulators are row-sums of 8 distinct M rows (all N
  // columns identical since B == ones). Lane pair (l, l^16) covers all 16 rows.
  float s = 0.f;
#pragma unroll
  for (int r = 0; r < 8; ++r) s += c[r];
  s += perm_x16(s);
  if (lane == 0) out[0] = s;
}

extern "C" void kernel_launch(void* const* d_in, const int* in_sizes, int n_in,
                              void* d_out, int out_size, void* d_ws, size_t ws_size,
                              hipStream_t stream) {
  (void)in_sizes; (void)n_in; (void)out_size; (void)ws_size;
  const float* obvs  = (const float*)d_in[0];  // [64, 1024]
  const float* mu    = (const float*)d_in[1];  // [512]
  const float* ln_pi = (const float*)d_in[2];  // [512]
  double* ws  = (double*)d_ws;                 // 64 per-batch results
  float*  out = (float*)d_out;                 // scalar fp32

  hmm_scan_kernel<<<HMM_B, 32, 0, stream>>>(obvs, mu, ln_pi, ws);
  hmm_reduce_kernel<<<1, 32, 0, stream>>>(ws, out);
}